// SL_CSC_IHT_43860206027563
// MI455X (gfx1250) — compile-verified
//
#include <hip/hip_runtime.h>
#include <hip/hip_bf16.h>
#include <stdint.h>

// Problem constants (from the reference)
#define NB   32
#define CA   128                 // dictionary atoms
#define CC   3                   // image channels
#define HX   57
#define WX   57
#define PP   (HX*WX)             // 3249 pixels of X
#define HY   64
#define WY   64
#define KH   8
#define KW   8
#define KDIM (CC*KH*KW)          // 192 = GEMM K for D^T
#define KSEL 2000
#define NPER (CA*PP)             // 415872 entries per sample

typedef float v2f __attribute__((ext_vector_type(2)));
typedef float v8f __attribute__((ext_vector_type(8)));

// ---------------------------------------------------------------------------
// Kernel 1: R = D(X) - Y   (transposed conv, only 3 output channels -> VALU)
// grid (16, 96): blockIdx.y = b*3+c, blockIdx.x tiles the 64x64 output.
// Weight slice W[:,c,:,:] (32 KB) staged in LDS.
// ---------------------------------------------------------------------------
__global__ void dres_kernel(const float* __restrict__ X, const float* __restrict__ Y,
                            const float* __restrict__ Wd, float* __restrict__ R) {
  const int bc = blockIdx.y;
  const int b = bc / 3, c = bc % 3;
  __shared__ float sw[CA * KH * KW];                // 32 KB
  for (int t = threadIdx.x; t < CA * 64; t += 256)
    sw[t] = Wd[((t >> 6) * 3 + c) * 64 + (t & 63)];
  __syncthreads();

  const int px = blockIdx.x * 256 + threadIdx.x;   // 0..4095
  const int oh = px >> 6, ow = px & 63;
  const int dh0 = (oh - (HX - 1)) > 0 ? (oh - (HX - 1)) : 0;
  const int dh1 = oh < (KH - 1) ? oh : (KH - 1);
  const int dw0 = (ow - (WX - 1)) > 0 ? (ow - (WX - 1)) : 0;
  const int dw1 = ow < (KW - 1) ? ow : (KW - 1);

  float acc = -Y[(bc * HY + oh) * WY + ow];
  for (int a = 0; a < CA; ++a) {
    const float* xa = X + ((size_t)(b * CA + a)) * PP;
    const float* wa = sw + a * 64;
    for (int dh = dh0; dh <= dh1; ++dh) {
      const float* xr = xa + (oh - dh) * WX + ow;
      const float* wr = wa + dh * KW;
      for (int dw = dw0; dw <= dw1; ++dw)
        acc = fmaf(xr[-dw], wr[dw], acc);
    }
  }
  R[(bc * HY + oh) * WY + ow] = acc;
}

// ---------------------------------------------------------------------------
// Kernel 2: HT = Xsrc - D^T(R)  via implicit-GEMM WMMA (fp32, 16x16x4).
// Each wave computes TWO 16x16 C tiles (32 atoms x 16 pixels): one B fetch
// feeds two WMMAs, halving global B traffic. 96 v_wmma per wave.
//
// Key codegen trick: for step ks (multiple of 4, koff in {0,2}) the B offset
//   c*HY*WY + kh*WY + kw  with  c=(ks>>6), kh=((ks&63)>>3), kw=(ks&7)+koff
// never crosses an octet/row/plane boundary, so c/kh/(ks&7) are compile-time
// constants after unroll and koff folds into the lane base pointer ->
// every B load becomes global_load_b64 with an immediate offset (no VALU).
//
// WMMA f32 16x16x4 VGPR layout assumed (ISA 7.12.2):
//   A: lane l -> row M=l%16; VGPR v -> K = 2*(l>=16) + v
//   B: lane l -> col N=l%16; VGPR v -> K = 2*(l>=16) + v
//   C: lane l -> N=l%16; VGPR r -> M = r + 8*(l>=16)
// grid (26, 4, 32): z=b, y=atom tile (32 atoms), x*8+wave = pixel tile.
// ---------------------------------------------------------------------------
__global__ void dt_wmma_kernel(const float* __restrict__ R, const float* __restrict__ Wd,
                               const float* __restrict__ Xsrc, float* __restrict__ HT) {
  const int b     = blockIdx.z;
  const int abase = blockIdx.y * 32;
  const int tid   = threadIdx.x;
  const int lane  = tid & 31;
  const int wv    = tid >> 5;

  __shared__ float As[32 * 196];                   // 25 KB, stride-196 pad
  // As[al][k] = W_D[abase+al, c, kw, kh] with k = c*64 + kh*8 + kw
  for (int t = tid; t < 32 * KDIM; t += 256) {
    const int al = t / KDIM, k = t - al * KDIM;
    const int c = k >> 6, rem = k & 63, kh = rem >> 3, kw = rem & 7;
    As[al * 196 + k] = Wd[(((abase + al) * 3 + c) * 8 + kw) * 8 + kh];
  }
  __syncthreads();

  const int ptile = blockIdx.x * 8 + wv;           // 0..207 (204 real tiles)
  const int p0    = ptile * 16;
  const int row16 = lane & 15;
  const int hi    = lane >> 4;                     // half-wave select
  const int koff  = hi * 2;

  int p = p0 + row16;                              // this lane's pixel column
  const int pcol = p;
  if (p > PP - 1) p = PP - 1;                      // clamp for safe gathers
  const int pi = p / WX, pj = p - pi * WX;

  // lane-constant bases; per-step displacements are immediates after unroll
  const float* lanebase = R + (size_t)b * CC * HY * WY + pi * WY + pj + koff;
  const float* arow0 = As + row16 * 196 + koff;
  const float* arow1 = arow0 + 16 * 196;

  v8f acc0 = {}, acc1 = {};
  #pragma unroll
  for (int ks = 0; ks < KDIM; ks += 4) {
    const int off = (ks >> 6) * (HY * WY) + ((ks & 63) >> 3) * WY + (ks & 7);
    v2f a0 = *(const v2f*)(arow0 + ks);            // 8B-aligned LDS reads
    v2f a1 = *(const v2f*)(arow1 + ks);
    v2f bv; bv.x = lanebase[off]; bv.y = lanebase[off + 1];
    acc0 = __builtin_amdgcn_wmma_f32_16x16x4_f32(
        false, a0, false, bv, (short)0, acc0, false, false);
    acc1 = __builtin_amdgcn_wmma_f32_16x16x4_f32(
        false, a1, false, bv, (short)0, acc1, false, false);
  }

  if (pcol < PP) {
    const size_t base = ((size_t)(b * CA + abase + hi * 8)) * PP + pcol;
    const float* xp = Xsrc + base;
    float*       hp = HT + base;
    #pragma unroll
    for (int r = 0; r < 8; ++r) {                  // immediate-offset, coalesced over lanes
      hp[(size_t)r * PP]        = xp[(size_t)r * PP]        - acc0[r];
      hp[(size_t)(r + 16) * PP] = xp[(size_t)(r + 16) * PP] - acc1[r];
    }
  }
}

// ---------------------------------------------------------------------------
// Kernel 3: exact per-sample radix select of the 2000th largest |value|.
// 4 passes x 256-bin LDS histogram over abs-value bits (monotone for floats).
// sel[s] = { threshold_key, ties_to_keep, tie_counter(=0) }
// ---------------------------------------------------------------------------
__global__ void select_kernel(const float* __restrict__ HT, uint32_t* __restrict__ sel) {
  const int s = blockIdx.x;
  const float* hs = HT + (size_t)s * NPER;
  __shared__ uint32_t hist[256];
  __shared__ uint32_t sh_prefix, sh_remk;
  if (threadIdx.x == 0) { sh_prefix = 0u; sh_remk = KSEL; }
  __syncthreads();

  for (int pass = 0; pass < 4; ++pass) {
    const int shift = 24 - 8 * pass;
    const uint32_t himask = (pass == 0) ? 0u : (0xFFFFFFFFu << (shift + 8));
    hist[threadIdx.x] = 0u;
    __syncthreads();
    const uint32_t pfx = sh_prefix;
    for (int i = threadIdx.x; i < NPER; i += 256) {
      const uint32_t key = __float_as_uint(hs[i]) & 0x7FFFFFFFu;
      if ((key & himask) == pfx)
        atomicAdd(&hist[(key >> shift) & 0xFF], 1u);
    }
    __syncthreads();
    if (threadIdx.x == 0) {
      uint32_t remk = sh_remk, cum = 0u;
      for (int bin = 255; bin >= 0; --bin) {
        const uint32_t c = hist[bin];
        if (cum + c >= remk) {
          sh_prefix = pfx | ((uint32_t)bin << shift);
          sh_remk = remk - cum;
          break;
        }
        cum += c;
      }
    }
    __syncthreads();
  }
  if (threadIdx.x == 0) {
    sel[s * 3 + 0] = sh_prefix;   // full 32-bit threshold key
    sel[s * 3 + 1] = sh_remk;     // number of threshold-equal entries to keep
    sel[s * 3 + 2] = 0u;          // tie claim counter (reset every iteration)
  }
}

// ---------------------------------------------------------------------------
// Kernel 4: scatter hard-thresholded result (exactly k kept per sample).
// grid (ceil(NPER/256), NB)
// ---------------------------------------------------------------------------
__global__ void scatter_kernel(const float* __restrict__ HT, uint32_t* __restrict__ sel,
                               float* __restrict__ out) {
  const int s = blockIdx.y;
  const int i = blockIdx.x * 256 + threadIdx.x;
  if (i >= NPER) return;
  const size_t idx = (size_t)s * NPER + i;
  const float v = HT[idx];
  const uint32_t key = __float_as_uint(v) & 0x7FFFFFFFu;
  const uint32_t T = sel[s * 3 + 0];
  float outv = 0.0f;
  if (key > T) {
    outv = v;
  } else if (key == T) {
    const uint32_t slot = atomicAdd(&sel[s * 3 + 2], 1u);
    if (slot < sel[s * 3 + 1]) outv = v;
  }
  out[idx] = outv;
}

// ---------------------------------------------------------------------------
// Launcher: 3 IHT iterations. d_out doubles as the running X buffer.
// Workspace: HT (53.2 MB) + R (1.57 MB) + sel (384 B)  ~= 54.8 MB.
// ---------------------------------------------------------------------------
extern "C" void kernel_launch(void* const* d_in, const int* in_sizes, int n_in,
                              void* d_out, int out_size, void* d_ws, size_t ws_size,
                              hipStream_t stream) {
  (void)in_sizes; (void)n_in; (void)out_size; (void)ws_size;
  const float* Y  = (const float*)d_in[0];
  const float* X0 = (const float*)d_in[1];
  const float* Wd = (const float*)d_in[2];
  float* out = (float*)d_out;

  char* ws = (char*)d_ws;
  float*    HT  = (float*)ws;                                        // 53,231,616 B
  float*    R   = (float*)(ws + (size_t)NB * NPER * sizeof(float));  // 1,572,864 B
  uint32_t* sel = (uint32_t*)(ws + (size_t)NB * NPER * sizeof(float)
                                 + (size_t)NB * CC * HY * WY * sizeof(float));

  const dim3 bdim(256);
  const dim3 gD(16, NB * CC);
  const dim3 gDt(26, 4, NB);        // 26*8 = 208 wave-tiles >= ceil(3249/16)=204
  const dim3 gSel(NB);
  const dim3 gSc((NPER + 255) / 256, NB);

  const float* Xs = X0;
  for (int it = 0; it < 3; ++it) {
    dres_kernel   <<<gD,   bdim, 0, stream>>>(Xs, Y, Wd, R);
    dt_wmma_kernel<<<gDt,  bdim, 0, stream>>>(R, Wd, Xs, HT);
    select_kernel <<<gSel, bdim, 0, stream>>>(HT, sel);
    scatter_kernel<<<gSc,  bdim, 0, stream>>>(HT, sel, out);
    Xs = out;
  }
}